// GCN_15590731285054
// MI455X (gfx1250) — compile-verified
//
#include <hip/hip_runtime.h>
#include <hip/hip_bf16.h>

typedef __attribute__((ext_vector_type(2))) float v2f;
typedef __attribute__((ext_vector_type(8))) float v8f;

#define DFEAT 128
#define A_STRIDE 132   // padded LDS stride (floats): bank = 4*m, conflict-free
#define GROWS 64       // rows per GEMM block

// ---------------- utility kernels ----------------

__global__ __launch_bounds__(256) void zero_f32(float* p, int n) {
    int i = blockIdx.x * 256 + threadIdx.x;
    if (i < n) p[i] = 0.0f;
}

__global__ __launch_bounds__(256) void degree_kernel(const int* __restrict__ src,
                                                     const int* __restrict__ dst,
                                                     float* __restrict__ deg_out,
                                                     float* __restrict__ deg_in, int E) {
    int e = blockIdx.x * 256 + threadIdx.x;
    if (e < E) {
        atomicAdd(&deg_out[src[e]], 1.0f);
        atomicAdd(&deg_in[dst[e]], 1.0f);
    }
}

// in-place deg -> rsqrt(max(deg,1)) over both arrays (contiguous, length 2N)
__global__ __launch_bounds__(256) void norm_kernel(float* deg, int n2) {
    int i = blockIdx.x * 256 + threadIdx.x;
    if (i < n2) {
        float d = deg[i];
        deg[i] = rsqrtf(d < 1.0f ? 1.0f : d);
    }
}

// xs[row, :] = h[row, :] * norm_out[row]; processed as float4s (n4 = N*32)
__global__ __launch_bounds__(256) void scale_kernel(const float* __restrict__ h,
                                                    const float* __restrict__ norm,
                                                    float* __restrict__ xs, int n4) {
    int i = blockIdx.x * 256 + threadIdx.x;
    if (i < n4) {
        int row = i >> 5;                 // 32 float4 per row
        float s = norm[row];
        float4 v = ((const float4*)h)[i];
        v.x *= s; v.y *= s; v.z *= s; v.w *= s;
        ((float4*)xs)[i] = v;
    }
}

// one wave (32 lanes) per edge; each lane moves 4 consecutive floats
__global__ __launch_bounds__(256) void scatter_add_kernel(const float* __restrict__ xs,
                                                          const int* __restrict__ src,
                                                          const int* __restrict__ dst,
                                                          float* __restrict__ agg, int E) {
    int e = blockIdx.x * 8 + (threadIdx.x >> 5);
    if (e >= E) return;
    int lane = threadIdx.x & 31;
    int s = src[e];
    int d = dst[e];
    const float4 v = *(const float4*)(xs + (size_t)s * DFEAT + lane * 4);
    float* p = agg + (size_t)d * DFEAT + lane * 4;
    atomicAdd(p + 0, v.x);
    atomicAdd(p + 1, v.y);
    atomicAdd(p + 2, v.z);
    atomicAdd(p + 3, v.w);
}

// ---------------- fused GEMM: out = relu((agg * norm_in) @ W + b) ----------------
// 256 threads = 8 waves; block computes GROWS rows x 128 cols.
// W staged in K-pair-interleaved LDS layout so every B fragment is one ds_load_b64.
// wave w: row-tile (w>>1), col-tiles (w&1)*4 .. +3; one A fragment feeds 4 parallel
// accumulator chains -> 128 v_wmma_f32_16x16x4_f32 per wave.
__global__ __launch_bounds__(256) void gemm_relu_kernel(const float* __restrict__ agg,
                                                        const float* __restrict__ norm_in,
                                                        const float* __restrict__ W,
                                                        const float* __restrict__ bias,
                                                        float* __restrict__ out, int n_nodes) {
    __shared__ float lA[GROWS * A_STRIDE];   // 33 KB, padded
    __shared__ float lWb[DFEAT * DFEAT];     // 64 KB, pair-interleaved:
                                             // lWb[((k>>1)*128 + n)*2 + (k&1)] = W[k][n]
    const int tid = threadIdx.x;
    const int row0 = blockIdx.x * GROWS;

    // stage W pair-interleaved: 8192 v2f, 32 per thread; coalesced global reads,
    // contiguous 64-bit LDS stores (conflict-free)
    {
        v2f* lW2 = (v2f*)lWb;
        #pragma unroll
        for (int i = tid; i < (DFEAT / 2) * DFEAT; i += 256) {
            int kp = i >> 7;                 // K pair index 0..63
            int n  = i & 127;
            v2f w;
            w.x = W[(2 * kp) * DFEAT + n];
            w.y = W[(2 * kp + 1) * DFEAT + n];
            lW2[i] = w;                      // offset (kp*128+n)*2
        }
    }
    // stage A tile, pre-scaled by norm_in (GROWS*32 float4, 8 per thread)
    {
        const float4* A4 = (const float4*)agg;
        #pragma unroll
        for (int i = tid; i < GROWS * (DFEAT / 4); i += 256) {
            int r  = i >> 5;                  // 0..GROWS-1
            int c4 = i & 31;                  // float4 index within row
            int row = row0 + r;
            float4 v;
            if (row < n_nodes) {
                float s = norm_in[row];
                v = A4[(size_t)row * (DFEAT / 4) + c4];
                v.x *= s; v.y *= s; v.z *= s; v.w *= s;
            } else {
                v.x = 0.f; v.y = 0.f; v.z = 0.f; v.w = 0.f;
            }
            *(float4*)&lA[r * A_STRIDE + c4 * 4] = v;
        }
    }
    __syncthreads();

    const int wave = tid >> 5;
    const int lane = tid & 31;
    const int m    = lane & 15;               // row (A) / col (B,C,D) within tile
    const int kk   = (lane >> 4) << 1;        // 0 or 2: K-pair selector
    const int half = lane >> 4;               // C/D row-half selector
    const int r0   = (wave >> 1) * 16;        // row tile base within block (0..48)
    const int c0   = (wave & 1) * 64;         // first col of this wave's 4 tiles

    v8f acc0 = {}, acc1 = {}, acc2 = {}, acc3 = {};
    const float* aRow = &lA[(r0 + m) * A_STRIDE];

    #pragma unroll 8
    for (int k0 = 0; k0 < DFEAT; k0 += 4) {
        // A 16x4 fragment: lane<16 -> K={k0,k0+1}, lane>=16 -> K={k0+2,k0+3}; one b64 load
        v2f a = *(const v2f*)&aRow[k0 + kk];
        // B 4x16 fragments: pair-interleaved -> one b64 load each
        const float* bRow = &lWb[((k0 + kk) >> 1) * (DFEAT * 2) + (c0 + m) * 2];
        v2f b0 = *(const v2f*)&bRow[0];
        v2f b1 = *(const v2f*)&bRow[32];
        v2f b2 = *(const v2f*)&bRow[64];
        v2f b3 = *(const v2f*)&bRow[96];
        acc0 = __builtin_amdgcn_wmma_f32_16x16x4_f32(false, a, false, b0, (short)0, acc0, false, false);
        acc1 = __builtin_amdgcn_wmma_f32_16x16x4_f32(false, a, false, b1, (short)0, acc1, false, false);
        acc2 = __builtin_amdgcn_wmma_f32_16x16x4_f32(false, a, false, b2, (short)0, acc2, false, false);
        acc3 = __builtin_amdgcn_wmma_f32_16x16x4_f32(false, a, false, b3, (short)0, acc3, false, false);
    }

    // C/D layout: VGPR j holds M = j + 8*half, N = m
    v8f accs[4] = { acc0, acc1, acc2, acc3 };
    #pragma unroll
    for (int t = 0; t < 4; ++t) {
        int c = c0 + t * 16 + m;
        float bv = bias[c];
        #pragma unroll
        for (int j = 0; j < 8; ++j) {
            int row = row0 + r0 + j + half * 8;
            if (row < n_nodes) {
                float v = accs[t][j] + bv;
                out[(size_t)row * DFEAT + c] = v > 0.0f ? v : 0.0f;
            }
        }
    }
}

// ---------------- mean readout ----------------

// each block (128 threads) sums 256 rows column-wise, atomically into acc[128]
__global__ __launch_bounds__(128) void colsum_kernel(const float* __restrict__ h,
                                                     float* __restrict__ acc, int n_nodes) {
    int t = threadIdx.x;
    int r0 = blockIdx.x * 256;
    int r1 = r0 + 256;
    if (r1 > n_nodes) r1 = n_nodes;
    float s = 0.0f;
    for (int r = r0; r < r1; ++r) s += h[(size_t)r * DFEAT + t];
    atomicAdd(&acc[t], s);
}

__global__ __launch_bounds__(128) void mean_finalize(const float* __restrict__ acc,
                                                     float* __restrict__ outmean, float invn) {
    int t = threadIdx.x;
    outmean[t] = acc[t] * invn;
}

// ---------------- launch ----------------

extern "C" void kernel_launch(void* const* d_in, const int* in_sizes, int n_in,
                              void* d_out, int out_size, void* d_ws, size_t ws_size,
                              hipStream_t stream) {
    const float* feature = (const float*)d_in[0];
    const int*   src     = (const int*)d_in[1];
    const int*   dst     = (const int*)d_in[2];
    const float* W1 = (const float*)d_in[3];
    const float* b1 = (const float*)d_in[4];
    const float* W2 = (const float*)d_in[5];
    const float* b2 = (const float*)d_in[6];
    const float* W3 = (const float*)d_in[7];
    const float* b3 = (const float*)d_in[8];

    const int N = in_sizes[0] / DFEAT;   // 100000
    const int E = in_sizes[1];           // 800000
    const size_t ND = (size_t)N * DFEAT;

    // workspace layout (floats): [norm_out N][norm_in N][acc 128][A ND][B ND]
    float* ws       = (float*)d_ws;
    float* norm_out = ws;
    float* norm_in  = ws + N;
    float* acc      = ws + 2 * (size_t)N;
    float* bufA     = acc + 128;
    float* bufB     = bufA + ND;

    float* h_out = (float*)d_out;            // first N*D elements: h
    float* hg    = h_out + ND;               // last 128: mean readout

    const int n4   = N * (DFEAT / 4);        // float4 count for N*D
    const int nd   = (int)ND;
    const int zb   = (nd + 255) / 256;
    const int z4b  = (n4 + 255) / 256;
    const int eb   = (E + 255) / 256;
    const int sb   = (E + 7) / 8;            // scatter: 8 edges per 256-thread block
    const int gb   = (N + GROWS - 1) / GROWS;
    const int cb   = (N + 255) / 256;

    // degrees -> norms
    zero_f32<<<(2 * N + 128 + 255) / 256, 256, 0, stream>>>(norm_out, 2 * N + 128);
    degree_kernel<<<eb, 256, 0, stream>>>(src, dst, norm_out, norm_in, E);
    norm_kernel<<<(2 * N + 255) / 256, 256, 0, stream>>>(norm_out, 2 * N);

    struct Layer { const float* I; float* X; float* G; float* O; const float* W; const float* b; };
    Layer layers[3] = {
        { feature, bufA, bufB, bufA, W1, b1 },
        { bufA,    bufB, bufA, bufB, W2, b2 },
        { bufB,    bufA, bufB, h_out, W3, b3 },
    };

    for (int l = 0; l < 3; ++l) {
        Layer& L = layers[l];
        scale_kernel<<<z4b, 256, 0, stream>>>(L.I, norm_out, L.X, n4);
        zero_f32<<<zb, 256, 0, stream>>>(L.G, nd);
        scatter_add_kernel<<<sb, 256, 0, stream>>>(L.X, src, dst, L.G, E);
        gemm_relu_kernel<<<gb, 256, 0, stream>>>(L.G, norm_in, L.W, L.b, L.O, N);
    }

    // mean readout (acc already zeroed above)
    colsum_kernel<<<cb, 128, 0, stream>>>(h_out, acc, N);
    mean_finalize<<<1, 128, 0, stream>>>(acc, hg, 1.0f / (float)N);
}